// GELU86_15255723835414
// MI455X (gfx1250) — compile-verified
//
#include <hip/hip_runtime.h>
#include <stdint.h>

// Problem shape (fixed by the reference).
#define B_   4
#define T_   4096
#define D_   2048
#define TS   64            // timesteps per segment
#define SEG  (T_ / TS)     // 64 segments
#define DC   256           // d-columns per block (= blockDim.x)
#define NDC  (D_ / DC)     // 8 column-chunks
#define NCH  (D_ / 32)     // 64 wave32-chunks across D

typedef uint32_t u32x4 __attribute__((ext_vector_type(4)));
typedef int      i32x4 __attribute__((ext_vector_type(4)));
typedef int      i32x8 __attribute__((ext_vector_type(8)));
typedef float    f32x4 __attribute__((ext_vector_type(4)));

__device__ __forceinline__ float gelu_tanh_f(float x) {
    const float c = 0.7978845608028654f;            // sqrt(2/pi)
    float t = tanhf(c * (x + 0.044715f * x * x * x));
    return 0.5f * x * (1.0f + t);
}

// wave32 butterfly sum (result in all lanes)
__device__ __forceinline__ float wave_sum(float v) {
#pragma unroll
    for (int off = 16; off > 0; off >>= 1) v += __shfl_xor(v, off, 32);
    return v;
}

// One TDM 2D tile load: tile_h rows x tile_w fp32 elements, row stride
// `stride_elems` (in elements), global base gsrc -> LDS byte offset lds_off.
// Descriptor layout per CDNA5 ISA sec. 8 (D# groups 0/1; groups 2/3 zero).
__device__ __forceinline__ void tdm_load_tile_f32(const float* gsrc, uint32_t lds_off,
                                                  uint32_t tile_w, uint32_t tile_h,
                                                  uint32_t stride_elems) {
    uint64_t ga = (uint64_t)(uintptr_t)gsrc;

    u32x4 g0;
    g0[0] = 1u;                                                   // count=1 (valid), user mode
    g0[1] = lds_off;                                              // lds_addr (bytes)
    g0[2] = (uint32_t)ga;                                         // global_addr[31:0]
    g0[3] = (uint32_t)((ga >> 32) & 0x01FFFFFFu) | (2u << 30);    // global_addr[56:32] | type=2

    i32x8 g1;
    g1[0] = (int)(2u << 16);                                      // workgroup_mask=0, data_size=2 (4B)
    g1[1] = (int)((tile_w & 0xFFFFu) << 16);                      // tensor_dim0[15:0]
    g1[2] = (int)((tile_w >> 16) | ((tile_h & 0xFFFFu) << 16));   // tensor_dim0[31:16] | tensor_dim1[15:0]
    g1[3] = (int)((tile_h >> 16) | ((tile_w & 0xFFFFu) << 16));   // tensor_dim1[31:16] | tile_dim0
    g1[4] = (int)(tile_h & 0xFFFFu);                              // tile_dim1 ; tile_dim2=0
    g1[5] = (int)stride_elems;                                    // tensor_dim0_stride[31:0]
    g1[6] = 0;                                                    // stride[47:32]=0 ; dim1_stride lo=0
    g1[7] = 0;

    i32x4 gz = {0, 0, 0, 0};
#if defined(__clang_major__) && (__clang_major__ >= 23)
    i32x8 gz8 = {0, 0, 0, 0, 0, 0, 0, 0};
    __builtin_amdgcn_tensor_load_to_lds(g0, g1, gz, gz, gz8, 0);
#else
    __builtin_amdgcn_tensor_load_to_lds(g0, g1, gz, gz, 0);
#endif
}

// ---------------- Pass 1: per-segment totals of x, x^2, gelu(x) ----------------
__global__ void k_seg_sums(const float* __restrict__ x,
                           float* __restrict__ ssx, float* __restrict__ ssxx,
                           float* __restrict__ sso) {
    __shared__ float tile[TS * DC];                 // 64 KB
    const int tid  = threadIdx.x;
    const int dblk = blockIdx.x, seg = blockIdx.y, b = blockIdx.z;
    const int d0   = dblk * DC;

    const float* src = x + ((size_t)b * T_ + (size_t)seg * TS) * D_ + d0;
    if ((tid >> 5) == 0) {                          // wave 0 issues the DMA (TDM ignores EXEC)
        tdm_load_tile_f32(src, (uint32_t)(uintptr_t)(void*)tile, DC, TS, D_);
        __builtin_amdgcn_s_wait_tensorcnt(0);
    }
    __syncthreads();

    float sx = 0.f, sxx = 0.f, so = 0.f;
#pragma unroll 4
    for (int i = 0; i < TS; ++i) {
        float xd = tile[i * DC + tid];
        sx  += xd;
        sxx += xd * xd;
        so  += gelu_tanh_f(xd);
    }
    size_t o = ((size_t)(b * SEG + seg)) * D_ + d0 + tid;
    ssx[o] = sx; ssxx[o] = sxx; sso[o] = so;
}

// ---------------- Pass 2: exclusive scan of segment totals (per column) ----------------
__global__ void k_scan_segs(float* __restrict__ ssx, float* __restrict__ ssxx,
                            float* __restrict__ sso) {
    const int d = blockIdx.x * blockDim.x + threadIdx.x;   // 0..D_-1
    const int b = blockIdx.z;
    float rx = 0.f, rxx = 0.f, ro = 0.f;
    for (int s = 0; s < SEG; ++s) {
        size_t o = ((size_t)(b * SEG + s)) * D_ + d;
        float vx = ssx[o], vxx = ssxx[o], vo = sso[o];
        ssx[o] = rx; ssxx[o] = rxx; sso[o] = ro;           // exclusive prefix
        rx += vx; rxx += vxx; ro += vo;
    }
}

// ---------------- Pass 3: in-segment scan + per-wave32 D-chunk partial reductions ----------------
__global__ void k_scan_main(const float* __restrict__ x,
                            const float* __restrict__ ssx, const float* __restrict__ ssxx,
                            const float* __restrict__ sso,
                            float4* __restrict__ partial) {
    __shared__ float tile[TS * DC];
    const int tid  = threadIdx.x;
    const int lane = tid & 31;
    const int wv   = tid >> 5;
    const int dblk = blockIdx.x, seg = blockIdx.y, b = blockIdx.z;
    const int d0   = dblk * DC;

    const float* src = x + ((size_t)b * T_ + (size_t)seg * TS) * D_ + d0;
    if (wv == 0) {
        tdm_load_tile_f32(src, (uint32_t)(uintptr_t)(void*)tile, DC, TS, D_);
        __builtin_amdgcn_s_wait_tensorcnt(0);
    }
    __syncthreads();

    size_t po = ((size_t)(b * SEG + seg)) * D_ + d0 + tid;
    float cx = ssx[po], cxx = ssxx[po], co = sso[po];       // exclusive carries
    const int chunk = dblk * (DC / 32) + wv;                // 0..NCH-1

    for (int i = 0; i < TS; ++i) {
        const int t   = seg * TS + i;
        float xd  = tile[i * DC + tid];
        float out = gelu_tanh_f(xd);

        float inv = 1.0f / (float)(t + 1);                  // counts[t] = t+1
        float mu  = cx  * inv;
        float sqm = cxx * inv;
        float var = fmaxf(sqm - mu * mu, 1e-4f);
        float z   = (xd - mu) / (sqrtf(var) + 1e-5f);
        float muo = co * inv;

        float a1 = wave_sum(fabsf(z));                      // sum |z|
        float a2 = wave_sum(out * out);                     // ||out||^2 partial
        float a3 = wave_sum(muo * muo);                     // ||mu_out||^2 partial
        float a4 = wave_sum(out * muo);                     // <out, mu_out> partial
        if (lane == 0)
            partial[((size_t)(b * T_ + t)) * NCH + chunk] = make_float4(a1, a2, a3, a4);

        cx += xd; cxx += xd * xd; co += out;                // advance (exclusive -> next)
    }
}

// ---------------- Pass 4: fold 64 chunk partials -> gate[b,t] ----------------
__global__ void k_gate(const float4* __restrict__ partial,
                       const float* __restrict__ p_log_tau,
                       const float* __restrict__ p_log_sigma_raw,
                       const float* __restrict__ p_log_w_raw,
                       float* __restrict__ gate) {
    const int lane = threadIdx.x & 31;
    const int wv   = threadIdx.x >> 5;
    const int bt   = blockIdx.x * 8 + wv;                   // one wave per (b,t)
    const int t    = bt & (T_ - 1);

    const float4 pa = partial[(size_t)bt * NCH + lane];
    const float4 pb = partial[(size_t)bt * NCH + lane + 32];
    float s1 = wave_sum(pa.x + pb.x);
    float s2 = wave_sum(pa.y + pb.y);
    float s3 = wave_sum(pa.z + pb.z);
    float s4 = wave_sum(pa.w + pb.w);

    if (lane == 0) {
        float tau   = expf(p_log_tau[0]);
        float sigma = log1pf(expf(p_log_sigma_raw[0]));     // softplus
        float w     = log1pf(expf(p_log_w_raw[0]));
        float mask  = (t >= 1) ? 1.0f : 0.0f;
        float maz   = (s1 * (1.0f / (float)D_)) * mask;
        float no    = fmaxf(sqrtf(s2), 1e-12f);
        float nm    = fmaxf(sqrtf(s3), 1e-12f);
        float cosv  = (s4 / (no * nm)) * mask;
        float surp  = tanhf(sigma * maz);
        gate[bt]    = expf(-tau * cosv) * (1.0f + w * surp);
    }
}

// ---------------- Pass 5: out = gelu(x) * gate (streaming, NT stores) ----------------
// Non-temporal stores keep x resident in the 192 MB L2 (x was warmed by pass 3),
// so the re-read of x here is an L2 hit and HBM sees ~read-once + write-once.
__global__ void k_apply(const f32x4* __restrict__ x4, const float* __restrict__ gate,
                        f32x4* __restrict__ o4) {
    const int n4 = B_ * T_ * D_ / 4;
    int i = blockIdx.x * blockDim.x + threadIdx.x;
    const int stride = gridDim.x * blockDim.x;
    for (; i < n4; i += stride) {
        f32x4 v = x4[i];
        float g = gate[i >> 9];                             // D_/4 = 512 float4 per row
        f32x4 r;
        r.x = gelu_tanh_f(v.x) * g;
        r.y = gelu_tanh_f(v.y) * g;
        r.z = gelu_tanh_f(v.z) * g;
        r.w = gelu_tanh_f(v.w) * g;
        __builtin_nontemporal_store(r, &o4[i]);             // TH=NT store
    }
}

extern "C" void kernel_launch(void* const* d_in, const int* in_sizes, int n_in,
                              void* d_out, int out_size, void* d_ws, size_t ws_size,
                              hipStream_t stream) {
    (void)in_sizes; (void)n_in; (void)out_size; (void)ws_size;
    const float* x  = (const float*)d_in[0];
    const float* lt = (const float*)d_in[1];
    const float* ls = (const float*)d_in[2];
    const float* lw = (const float*)d_in[3];

    // Workspace layout (needs ~22.1 MB):
    //   [0,2MB)   ssx   : [B,SEG,D] f32
    //   [2,4MB)   ssxx
    //   [4,6MB)   sso
    //   [6,22MB)  partial: [B,T,NCH] float4
    //   [22MB,+64KB) gate: [B,T] f32
    char*   ws      = (char*)d_ws;
    float*  ssx     = (float*)(ws);
    float*  ssxx    = (float*)(ws + (size_t)2 * 1024 * 1024);
    float*  sso     = (float*)(ws + (size_t)4 * 1024 * 1024);
    float4* partial = (float4*)(ws + (size_t)6 * 1024 * 1024);
    float*  gate    = (float*)(ws + (size_t)22 * 1024 * 1024);

    k_seg_sums <<<dim3(NDC, SEG, B_), DC, 0, stream>>>(x, ssx, ssxx, sso);
    k_scan_segs<<<dim3(NDC, 1, B_),   DC, 0, stream>>>(ssx, ssxx, sso);
    k_scan_main<<<dim3(NDC, SEG, B_), DC, 0, stream>>>(x, ssx, ssxx, sso, partial);
    k_gate     <<<dim3((B_ * T_) / 8), 256, 0, stream>>>(partial, lt, ls, lw, gate);
    k_apply    <<<dim3(4096), 256, 0, stream>>>((const f32x4*)x, gate, (f32x4*)d_out);
}